// TransformerBlock_35416300323762
// MI455X (gfx1250) — compile-verified
//
#include <hip/hip_runtime.h>
#include <math.h>

// Problem constants (match reference)
constexpr int kN  = 50000;
constexpr int kE  = 800000;
constexpr int kIN = 64;
constexpr int kC  = 32;
constexpr int kH  = 4;
constexpr int kHC = 128;   // H*C
constexpr int kB  = 64;
constexpr float kEPS = 1e-5f;
constexpr float kInvSqrtC = 0.17677669529663687f; // 1/sqrt(32)

typedef float v2f __attribute__((ext_vector_type(2)));
typedef float v8f __attribute__((ext_vector_type(8)));

using idx_t = int;

// ---- gfx1250 async global->LDS path (guarded; falls back to plain copy) ----
#if defined(__has_builtin)
#  if __has_builtin(__builtin_amdgcn_global_load_async_to_lds_b128) && \
      __has_builtin(__builtin_amdgcn_s_wait_asynccnt)
#    define USE_ASYNC_LDS 1
#  endif
#endif
#ifndef USE_ASYNC_LDS
#  define USE_ASYNC_LDS 0
#endif

// Builtin expects: (int4 AS1*, int4 AS3*, imm offset, imm cpol)
typedef int v4i __attribute__((vector_size(4 * sizeof(int))));
typedef __attribute__((address_space(1))) v4i g_v4i;
typedef __attribute__((address_space(3))) v4i l_v4i;

// Stage a 16 x kw fp32 tile (contiguous rows in global) into LDS with a
// padded row stride (bank-conflict free for the column-wise A-fragment reads).
__device__ __forceinline__ void stage_tile16(const float* __restrict__ gsrc,
                                             float* lsdst, int ldsStride,
                                             int kw, int nthreads, int t)
{
  const int chunksPerRow = kw >> 2;          // float4 chunks
  const int total = 16 * chunksPerRow;
#if USE_ASYNC_LDS
  for (int i = t; i < total; i += nthreads) {
    const int row = i / chunksPerRow;
    const int c4  = i % chunksPerRow;
    __builtin_amdgcn_global_load_async_to_lds_b128(
        (g_v4i*)(gsrc + (size_t)row * kw + c4 * 4),
        (l_v4i*)(lsdst + row * ldsStride + c4 * 4),
        0, 0);
  }
  __builtin_amdgcn_s_wait_asynccnt(0);   // drain this wave's async copies
  __syncthreads();                       // all waves' copies visible
#else
  for (int i = t; i < total; i += nthreads) {
    const int row = i / chunksPerRow;
    const int c4  = i % chunksPerRow;
    const float4 d = *(const float4*)(gsrc + (size_t)row * kw + c4 * 4);
    *(float4*)(lsdst + row * ldsStride + c4 * 4) = d;
  }
  __syncthreads();
#endif
}

// ---------- helpers ----------
__device__ __forceinline__ unsigned f2ord(float f) {
  unsigned u = __float_as_uint(f);
  return (u & 0x80000000u) ? ~u : (u | 0x80000000u);
}
__device__ __forceinline__ float ord2f(unsigned u) {
  return __uint_as_float((u & 0x80000000u) ? (u ^ 0x80000000u) : ~u);
}

// ---------- zero scratch accumulators ----------
__global__ void zero_kernel(float* __restrict__ p, long long n) {
  long long i = (long long)blockIdx.x * blockDim.x + threadIdx.x;
  long long stride = (long long)gridDim.x * blockDim.x;
  for (; i < n; i += stride) p[i] = 0.0f;
}

// ---------- K1: fused Q/K/V/Skip projection, fp32 WMMA 16x16x4 ----------
// grid = (N/16, 8), block = 128 (4 waves). All waves share one 16x64 A-tile
// staged in LDS (async copy); wave w computes one 16x16 tile of the virtual
// [N,512] = x @ [Wq|Wk|Wv|Wskip] GEMM.
constexpr int kXStride = kIN + 4;   // 68: float4-aligned, stride%64==4 -> no bank conflicts
__global__ __launch_bounds__(128) void proj_wmma(
    const float* __restrict__ x,
    const float* __restrict__ Wq, const float* __restrict__ bq,
    const float* __restrict__ Wk, const float* __restrict__ bk,
    const float* __restrict__ Wv, const float* __restrict__ bv,
    const float* __restrict__ Wsk, const float* __restrict__ bsk,
    float* __restrict__ q, float* __restrict__ k,
    float* __restrict__ v, float* __restrict__ s)
{
  __shared__ float xs[16 * kXStride];
  const int lane = threadIdx.x & 31;
  const int wave = threadIdx.x >> 5;
  const int hl   = lane >> 4;       // half-wave: 0 or 1
  const int l16  = lane & 15;
  const int m0   = blockIdx.x * 16;
  const int col  = blockIdx.y * 64 + wave * 16;   // 0..511
  const int mat  = col >> 7;
  const int n0   = col & 127;

  stage_tile16(x + (size_t)m0 * kIN, xs, kXStride, kIN, 128, threadIdx.x);

  const float* W; const float* bias; float* out;
  switch (mat) {
    case 0:  W = Wq;  bias = bq;  out = q; break;
    case 1:  W = Wk;  bias = bk;  out = k; break;
    case 2:  W = Wv;  bias = bv;  out = v; break;
    default: W = Wsk; bias = bsk; out = s; break;
  }

  const float* xrow = xs + l16 * kXStride;
  v8f acc = {0.f,0.f,0.f,0.f,0.f,0.f,0.f,0.f};
#pragma unroll
  for (int k0 = 0; k0 < kIN; k0 += 4) {
    // A fragment from LDS: A[M=l16][K = k0 + 2*hl + {0,1}]
    const float2 av = *(const float2*)(xrow + k0 + 2 * hl);
    v2f a; a.x = av.x; a.y = av.y;
    // B fragment: B[K = k0 + 2*hl + {0,1}][N = n0 + l16]
    v2f b;
    b.x = W[(size_t)(k0 + 2 * hl)     * kHC + n0 + l16];
    b.y = W[(size_t)(k0 + 2 * hl + 1) * kHC + n0 + l16];
    acc = __builtin_amdgcn_wmma_f32_16x16x4_f32(false, a, false, b,
                                                (short)0, acc, false, false);
  }
  const float bn = bias[n0 + l16];
#pragma unroll
  for (int i = 0; i < 8; ++i) {
    const int row = m0 + i + 8 * hl;   // C/D layout: M = vgpr + 8*half
    out[(size_t)row * kHC + n0 + l16] = acc[i] + bn;
  }
}

// ---------- K2a: per-edge attention logits + segment max ----------
__global__ __launch_bounds__(256) void edge_logits(
    const float* __restrict__ q, const float* __restrict__ k,
    const idx_t* __restrict__ ei, const float* __restrict__ ea,
    const float* __restrict__ We, const float* __restrict__ be,
    float* __restrict__ alpha, unsigned* __restrict__ mmax)
{
  const long long t = (long long)blockIdx.x * blockDim.x + threadIdx.x;
  if (t >= (long long)kE * kH) return;
  const int e = (int)(t >> 2);
  const int h = (int)(t & 3);
  const idx_t src = ei[e];
  const idx_t dst = ei[kE + e];
  const float eav = ea[e];
  const float* qrow = q + (size_t)dst * kHC + h * kC;
  const float* krow = k + (size_t)src * kHC + h * kC;
  float acc = 0.f;
#pragma unroll
  for (int c = 0; c < kC; ++c) {
    const float ke = krow[c] + (eav * We[h * kC + c] + be[h * kC + c]);
    acc = fmaf(qrow[c], ke, acc);
  }
  acc *= kInvSqrtC;
  alpha[t] = acc;
  atomicMax(&mmax[(size_t)dst * kH + h], f2ord(acc));
}

// ---------- K2b: exp(alpha - max) + segment sum ----------
__global__ __launch_bounds__(256) void edge_exp(
    const idx_t* __restrict__ ei, const unsigned* __restrict__ mmax,
    float* __restrict__ alpha, float* __restrict__ denom)
{
  const long long t = (long long)blockIdx.x * blockDim.x + threadIdx.x;
  if (t >= (long long)kE * kH) return;
  const int e = (int)(t >> 2);
  const int h = (int)(t & 3);
  const idx_t dst = ei[kE + e];
  const float m = ord2f(mmax[(size_t)dst * kH + h]);
  const float a = expf(alpha[t] - m);
  alpha[t] = a;                      // reuse buffer as 'a'
  atomicAdd(&denom[(size_t)dst * kH + h], a);
}

// ---------- K2c: weighted message aggregation ----------
// One block (128 threads) per edge; thread t covers channel hc=t, head h=t>>5.
__global__ __launch_bounds__(128) void edge_aggregate(
    const idx_t* __restrict__ ei, const float* __restrict__ ea,
    const float* __restrict__ We, const float* __restrict__ be,
    const float* __restrict__ v, const float* __restrict__ a,
    const float* __restrict__ denom, float* __restrict__ agg)
{
  const int e = blockIdx.x;
  const int t = threadIdx.x;
  const int h = t >> 5;
  const idx_t src = ei[e];
  const idx_t dst = ei[kE + e];
  const float coef = a[(size_t)e * kH + h] /
                     fmaxf(denom[(size_t)dst * kH + h], 1e-16f);
  const float msg = (v[(size_t)src * kHC + t] + (ea[e] * We[t] + be[t])) * coef;
  atomicAdd(&agg[(size_t)dst * kHC + t], msg);
}

// ---------- K3a: beta-gated skip ----------
__global__ __launch_bounds__(128) void beta_gate(
    const float* __restrict__ agg, const float* __restrict__ sk,
    const float* __restrict__ Wbeta, float* __restrict__ gated)
{
  __shared__ float red[128];
  const int n = blockIdx.x;
  const int t = threadIdx.x;
  const float o  = agg[(size_t)n * kHC + t];
  const float xr = sk[(size_t)n * kHC + t];
  red[t] = o * Wbeta[t] + xr * Wbeta[kHC + t] + (o - xr) * Wbeta[2 * kHC + t];
  __syncthreads();
  for (int off = 64; off > 0; off >>= 1) {
    if (t < off) red[t] += red[t + off];
    __syncthreads();
  }
  const float beta = 1.f / (1.f + expf(-red[0]));
  gated[(size_t)n * kHC + t] = beta * xr + (1.f - beta) * o;
}

// ---------- K3b: h = relu(gated @ Wlin + blin), fp32 WMMA, K=128 ----------
// grid = N/16, block = 64 (2 waves) sharing one 16x128 A-tile in LDS;
// wave w -> 16-col tile of the 32 output cols.
constexpr int kGStride = kHC + 4;   // 132: float4-aligned, stride%64==4
__global__ __launch_bounds__(64) void lin_relu_wmma(
    const float* __restrict__ gated, const float* __restrict__ Wlin,
    const float* __restrict__ blin, float* __restrict__ hout)
{
  __shared__ float gs[16 * kGStride];
  const int lane = threadIdx.x & 31;
  const int wave = threadIdx.x >> 5;
  const int hl   = lane >> 4;
  const int l16  = lane & 15;
  const int m0   = blockIdx.x * 16;
  const int n0   = wave * 16;

  stage_tile16(gated + (size_t)m0 * kHC, gs, kGStride, kHC, 64, threadIdx.x);

  const float* grow = gs + l16 * kGStride;
  v8f acc = {0.f,0.f,0.f,0.f,0.f,0.f,0.f,0.f};
#pragma unroll
  for (int k0 = 0; k0 < kHC; k0 += 4) {
    const float2 av = *(const float2*)(grow + k0 + 2 * hl);
    v2f a; a.x = av.x; a.y = av.y;
    v2f b;
    b.x = Wlin[(size_t)(k0 + 2 * hl)     * kC + n0 + l16];
    b.y = Wlin[(size_t)(k0 + 2 * hl + 1) * kC + n0 + l16];
    acc = __builtin_amdgcn_wmma_f32_16x16x4_f32(false, a, false, b,
                                                (short)0, acc, false, false);
  }
  const float bn = blin[n0 + l16];
#pragma unroll
  for (int i = 0; i < 8; ++i) {
    const int row = m0 + i + 8 * hl;
    hout[(size_t)row * kC + n0 + l16] = fmaxf(acc[i] + bn, 0.f);
  }
}

// ---------- K4: GraphNorm ----------
__global__ __launch_bounds__(32) void gn_sum(
    const float* __restrict__ h, const idx_t* __restrict__ bidx,
    float* __restrict__ gsum, float* __restrict__ gcnt)
{
  const int n = blockIdx.x;
  const int c = threadIdx.x;
  const idx_t b = bidx[n];
  atomicAdd(&gsum[b * kC + c], h[(size_t)n * kC + c]);
  if (c == 0) atomicAdd(&gcnt[b], 1.0f);
}

__global__ void gn_mean(const float* __restrict__ gsum,
                        const float* __restrict__ gcnt,
                        float* __restrict__ mean)
{
  const int i = blockIdx.x * blockDim.x + threadIdx.x;
  if (i >= kB * kC) return;
  mean[i] = gsum[i] / fmaxf(gcnt[i / kC], 1.0f);
}

__global__ __launch_bounds__(32) void gn_var(
    const float* __restrict__ h, const idx_t* __restrict__ bidx,
    const float* __restrict__ mean, const float* __restrict__ msc,
    float* __restrict__ gvar)
{
  const int n = blockIdx.x;
  const int c = threadIdx.x;
  const idx_t b = bidx[n];
  const float cen = h[(size_t)n * kC + c] - mean[b * kC + c] * msc[c];
  atomicAdd(&gvar[b * kC + c], cen * cen);
}

__global__ __launch_bounds__(32) void gn_out(
    const float* __restrict__ h, const idx_t* __restrict__ bidx,
    const float* __restrict__ mean, const float* __restrict__ gvar,
    const float* __restrict__ gcnt, const float* __restrict__ msc,
    const float* __restrict__ gw, const float* __restrict__ gb,
    float* __restrict__ out)
{
  const int n = blockIdx.x;
  const int c = threadIdx.x;
  const idx_t b = bidx[n];
  const float cen = h[(size_t)n * kC + c] - mean[b * kC + c] * msc[c];
  const float var = gvar[b * kC + c] / fmaxf(gcnt[b], 1.0f);
  out[(size_t)n * kC + c] = gw[c] * cen / sqrtf(var + kEPS) + gb[c];
}

// ---------- launch ----------
extern "C" void kernel_launch(void* const* d_in, const int* in_sizes, int n_in,
                              void* d_out, int out_size, void* d_ws, size_t ws_size,
                              hipStream_t stream)
{
  const float* x    = (const float*)d_in[0];
  const idx_t* ei   = (const idx_t*)d_in[1];
  const float* ea   = (const float*)d_in[2];
  const idx_t* bidx = (const idx_t*)d_in[3];
  const float* Wq = (const float*)d_in[4];   const float* bq = (const float*)d_in[5];
  const float* Wk = (const float*)d_in[6];   const float* bk = (const float*)d_in[7];
  const float* Wv = (const float*)d_in[8];   const float* bv = (const float*)d_in[9];
  const float* We = (const float*)d_in[10];  const float* be = (const float*)d_in[11];
  const float* Wsk= (const float*)d_in[12];  const float* bsk= (const float*)d_in[13];
  const float* Wbeta = (const float*)d_in[14];
  const float* Wlin  = (const float*)d_in[15];
  const float* blin  = (const float*)d_in[16];
  const float* gw  = (const float*)d_in[17];
  const float* gb  = (const float*)d_in[18];
  const float* gms = (const float*)d_in[19];
  float* out = (float*)d_out;

  float* ws = (float*)d_ws;
  size_t off = 0;
  float* q     = ws + off; off += (size_t)kN * kHC;
  float* k     = ws + off; off += (size_t)kN * kHC;
  float* v     = ws + off; off += (size_t)kN * kHC;
  float* s     = ws + off; off += (size_t)kN * kHC;
  float* alpha = ws + off; off += (size_t)kE * kH;   // reused as 'a'
  const size_t zstart = off;                          // zeroed region starts here
  unsigned* mmax = (unsigned*)(ws + off); off += (size_t)kN * kH;
  float* denom = ws + off; off += (size_t)kN * kH;
  float* agg   = ws + off; off += (size_t)kN * kHC;
  float* gsum  = ws + off; off += (size_t)kB * kC;
  float* gvar  = ws + off; off += (size_t)kB * kC;
  float* gcnt  = ws + off; off += (size_t)kB;
  float* mean  = ws + off; off += (size_t)kB * kC;
  const long long zcount = (long long)(off - zstart);
  float* gated = q;   // q is dead after edge_logits
  float* hbuf  = k;   // k is dead after edge_logits

  zero_kernel<<<2048, 256, 0, stream>>>(ws + zstart, zcount);

  dim3 g1(kN / 16, 8);
  proj_wmma<<<g1, 128, 0, stream>>>(x, Wq, bq, Wk, bk, Wv, bv, Wsk, bsk, q, k, v, s);

  const long long eh = (long long)kE * kH;
  edge_logits<<<(unsigned)((eh + 255) / 256), 256, 0, stream>>>(q, k, ei, ea, We, be, alpha, mmax);
  edge_exp<<<(unsigned)((eh + 255) / 256), 256, 0, stream>>>(ei, mmax, alpha, denom);
  edge_aggregate<<<kE, 128, 0, stream>>>(ei, ea, We, be, v, alpha, denom, agg);

  beta_gate<<<kN, 128, 0, stream>>>(agg, s, Wbeta, gated);
  lin_relu_wmma<<<kN / 16, 64, 0, stream>>>(gated, Wlin, blin, hbuf);

  gn_sum<<<kN, 32, 0, stream>>>(hbuf, bidx, gsum, gcnt);
  gn_mean<<<(kB * kC + 255) / 256, 256, 0, stream>>>(gsum, gcnt, mean);
  gn_var<<<kN, 32, 0, stream>>>(hbuf, bidx, mean, gms, gvar);
  gn_out<<<kN, 32, 0, stream>>>(hbuf, bidx, mean, gvar, gcnt, gms, gw, gb, out);
}